// _SmoothMatrixDecompositionBase_83889301225943
// MI455X (gfx1250) — compile-verified
//
#include <hip/hip_runtime.h>
#include <math.h>

typedef __attribute__((ext_vector_type(2))) float v2f;
typedef __attribute__((ext_vector_type(8))) float v8f;

#define D_      160
#define N_      32768
#define NB_     8
#define K_      589
#define R_      8
#define EPS_    1e-6f
#define TILE_   128
#define NTILES_ 256          // N_/TILE_
#define PITCH_  130          // padded LDS row pitch for x tile (floats)
#define EBP_    17           // padded pitch for 16-wide zero-padded Eb / coef tiles

// workspace offsets (in floats)
#define O_E      0                              // 160*589   = 94240
#define O_BASES  94240                          // 8*589*8   = 37696
#define O_EB     131936                         // 8*160*8   = 10240
#define O_ETE    142176                         // 8*64      = 512
#define O_NUMB   142688                         // 37696
#define O_BC     180384                         // 37696
#define O_T2     218080                         // 10240
#define O_XC     228320                         // 10240
#define O_CTC    238560                         // 512
#define O_COEF   239072                         // 8*32768*8 = 2097152
#define O_PXC    2336224                        // 8*256*1280 = 2621440
#define O_PCTC   4957664                        // 8*256*64   = 131072

// ---------------- small setup kernels ----------------

__global__ void k_build_E(float* __restrict__ E) {
  int idx = blockIdx.x * 256 + threadIdx.x;
  if (idx >= D_ * K_) return;
  int t = idx / K_, col = idx % K_;
  float sig, n;
  if      (col < 160) { sig = 6.f;  n = (float)col; }
  else if (col < 320) { sig = 8.f;  n = (float)(col - 160); }
  else if (col < 400) { sig = 12.f; n = (float)((col - 320) * 2); }
  else if (col < 480) { sig = 15.f; n = (float)((col - 400) * 2); }
  else if (col < 534) { sig = 18.f; n = (float)((col - 480) * 3); }
  else if (col < 588) { sig = 24.f; n = (float)((col - 534) * 3); }
  else { E[idx] = 1.f; return; }
  float d = (float)t - n;
  E[idx] = expf(-0.5f * (d * d) / (2.f * sig * sig));
}

__global__ void k_init_bases(const float* __restrict__ bi, float* __restrict__ bases) {
  int b = blockIdx.x;
  int r = threadIdx.x >> 5, lane = threadIdx.x & 31;   // 8 waves, wave==column r
  const float* src = bi + (size_t)b * K_ * R_;
  float* dst = bases + (size_t)b * K_ * R_;
  float s = 0.f;
  for (int k = lane; k < K_; k += 32) { float v = src[k * R_ + r]; s += v * v; }
  for (int off = 16; off; off >>= 1) s += __shfl_xor(s, off, 32);
  float inv = 1.f / fmaxf(sqrtf(s), 1e-12f);
  for (int k = lane; k < K_; k += 32) dst[k * R_ + r] = src[k * R_ + r] * inv;
}

__global__ void k_Eb(const float* __restrict__ E, const float* __restrict__ bases,
                     float* __restrict__ Eb) {
  int b = blockIdx.y;
  int e = blockIdx.x * 256 + threadIdx.x;              // 5*256 == 1280 exactly
  int d = e >> 3, r = e & 7;
  const float* Er = E + (size_t)d * K_;
  const float* bb = bases + (size_t)b * K_ * R_ + r;
  float s = 0.f;
  for (int k = 0; k < K_; ++k) s += Er[k] * bb[(size_t)k * R_];
  Eb[(size_t)b * D_ * R_ + e] = s;
}

__global__ void k_EtE(const float* __restrict__ Eb, float* __restrict__ EtE) {
  int b = blockIdx.x, t = threadIdx.x;                 // 64 threads
  int r = t >> 3, s = t & 7;
  const float* e = Eb + (size_t)b * D_ * R_;
  float acc = 0.f;
  for (int d = 0; d < D_; ++d) acc += e[d * R_ + r] * e[d * R_ + s];
  EtE[b * 64 + t] = acc;
}

// ---------------- per-step bases-update chain ----------------

__global__ void k_reduce(const float* __restrict__ pxc, const float* __restrict__ pctc,
                         float* __restrict__ xc, float* __restrict__ ctc) {
  int b = blockIdx.y;
  if (blockIdx.x < 5) {
    int e = blockIdx.x * 256 + threadIdx.x;            // < 1280
    const float* p = pxc + (size_t)b * NTILES_ * (D_ * R_) + e;
    float s = 0.f;
    for (int t = 0; t < NTILES_; ++t) s += p[(size_t)t * (D_ * R_)];
    xc[(size_t)b * D_ * R_ + e] = s;
  } else if (threadIdx.x < 64) {
    const float* p = pctc + (size_t)b * NTILES_ * 64 + threadIdx.x;
    float s = 0.f;
    for (int t = 0; t < NTILES_; ++t) s += p[t * 64];
    ctc[b * 64 + threadIdx.x] = s;
  }
}

__global__ void k_numb_bc(const float* __restrict__ E, const float* __restrict__ xc,
                          const float* __restrict__ bases, const float* __restrict__ ctc,
                          float* __restrict__ numb, float* __restrict__ bc) {
  int b = blockIdx.y;
  int e = blockIdx.x * 256 + threadIdx.x;
  if (e >= K_ * R_) return;
  int k = e >> 3, r = e & 7;
  const float* xcb = xc + (size_t)b * D_ * R_;
  float s = 0.f;
  for (int d = 0; d < D_; ++d) s += E[(size_t)d * K_ + k] * xcb[d * R_ + r];
  numb[(size_t)b * K_ * R_ + e] = s;
  const float* bb = bases + (size_t)b * K_ * R_ + (size_t)k * R_;
  const float* cb = ctc + b * 64;
  float t = 0.f;
  for (int q = 0; q < R_; ++q) t += bb[q] * cb[q * R_ + r];
  bc[(size_t)b * K_ * R_ + e] = t;
}

__global__ void k_t2(const float* __restrict__ E, const float* __restrict__ bc,
                     float* __restrict__ t2) {
  int b = blockIdx.y;
  int e = blockIdx.x * 256 + threadIdx.x;              // 1280 exactly
  int d = e >> 3, r = e & 7;
  const float* Er = E + (size_t)d * K_;
  const float* bb = bc + (size_t)b * K_ * R_ + r;
  float s = 0.f;
  for (int k = 0; k < K_; ++k) s += Er[k] * bb[(size_t)k * R_];
  t2[(size_t)b * D_ * R_ + e] = s;
}

__global__ void k_denb(const float* __restrict__ E, const float* __restrict__ t2,
                       const float* __restrict__ numb, float* __restrict__ bases) {
  int b = blockIdx.y;
  int e = blockIdx.x * 256 + threadIdx.x;
  if (e >= K_ * R_) return;
  int k = e >> 3, r = e & 7;
  const float* tb = t2 + (size_t)b * D_ * R_;
  float den = 0.f;
  for (int d = 0; d < D_; ++d) den += E[(size_t)d * K_ + k] * tb[d * R_ + r];
  size_t idx = (size_t)b * K_ * R_ + e;
  bases[idx] = bases[idx] * numb[idx] / (den + EPS_);
}

// ---------------- the fused big pass (one of 5 passes over x) ----------------
// mode 0: coef = softmax(x^T Eb); coef update (step 1); accumulate xc,ctc
// mode 1: coef update (steps 2..4); accumulate xc,ctc
// mode 2: compute_coef + reconstruction to d_out

__global__ __launch_bounds__(256) void k_pass(
    const float* __restrict__ x, const float* __restrict__ Eb_g,
    const float* __restrict__ EtE_g, float* __restrict__ coef_g,
    float* __restrict__ pxc, float* __restrict__ pctc,
    float* __restrict__ out, int mode) {
  __shared__ float lx[D_ * PITCH_];      // 83.2 KB: x tile [160 x 128], padded rows
  __shared__ float lEb[D_ * EBP_];       // zero-padded to 16 cols (cols 8..15 == 0)
  __shared__ float lEtE[R_ * R_];
  __shared__ float lcoef[8][16 * EBP_];  // per-wave 16x16 coef tile, zero-padded
  __shared__ float lxc[D_ * R_];
  __shared__ float lctc[R_ * R_];

  const int tid  = threadIdx.x;
  const int wave = tid >> 5;
  const int lane = tid & 31;
  const int l    = lane & 15;
  const int hi   = lane >> 4;
  const int b    = blockIdx.y;
  const int tile0 = blockIdx.x * TILE_;

  // zero-padded Eb tile: data padding instead of EXEC-masked loads
  for (int i = tid; i < D_ * 16; i += 256) {
    int row = i >> 4, c = i & 15;
    lEb[row * EBP_ + c] = (c < R_) ? Eb_g[b * D_ * R_ + row * R_ + c] : 0.f;
  }
  for (int i = tid; i < 8 * 16 * EBP_; i += 256) (&lcoef[0][0])[i] = 0.f;
  if (tid < 64) lEtE[tid] = EtE_g[b * 64 + tid];
  if (mode <= 1) {
    for (int i = tid; i < D_ * R_; i += 256) lxc[i] = 0.f;
    if (tid < 64) lctc[tid] = 0.f;
  }

  // ---- async stage x tile (160 rows x 512B) into LDS ----
  {
    const float* xb = x + (size_t)b * (size_t)(D_ * N_);
    for (int i = 0; i < 20; ++i) {
      int li = i * 256 + tid;
      int d = li >> 5, seg = li & 31;
      unsigned loff = (unsigned)(size_t)(&lx[d * PITCH_ + seg * 4]);
      unsigned long long ga =
          (unsigned long long)(size_t)(xb + (size_t)d * N_ + tile0 + seg * 4);
      asm volatile("global_load_async_to_lds_b128 %0, %1, off"
                   :: "v"(loff), "v"(ga) : "memory");
    }
    asm volatile("s_wait_asynccnt 0x0" ::: "memory");
  }
  __syncthreads();

  const int col16 = wave * 16;

  // ---- Phase A: num[r][n] = sum_d Eb[d][r] * x[d][n], WMMA f32 16x16x4, K=d ----
  v8f num = {};
#pragma unroll 4
  for (int dc = 0; dc < 40; ++dc) {
    int d0 = dc * 4 + hi * 2;
    v2f a, bb;
    a[0] = lEb[d0 * EBP_ + l];                         // A = Eb^T (rows r, 0-padded)
    a[1] = lEb[(d0 + 1) * EBP_ + l];
    bb[0] = lx[d0 * PITCH_ + col16 + l];               // B = x chunk
    bb[1] = lx[(d0 + 1) * PITCH_ + col16 + l];
    num = __builtin_amdgcn_wmma_f32_16x16x4_f32(false, a, false, bb, (short)0,
                                                num, false, false);
  }
  // lane n (<16) now holds num[0..7][n] across the 8 accumulator components.

  // ---- Phase B: per-column coef update ----
  float co[R_], cn[R_];
  const size_t cbase = ((size_t)b * N_ + (size_t)(tile0 + col16 + l)) * R_;
  if (mode == 0) {
    float m = num[0];
    for (int i = 1; i < R_; ++i) m = fmaxf(m, num[i]);
    float s = 0.f;
    for (int i = 0; i < R_; ++i) { co[i] = expf(num[i] - m); s += co[i]; }
    float inv = 1.f / s;
    for (int i = 0; i < R_; ++i) co[i] *= inv;
  } else {
    for (int i = 0; i < R_; ++i) co[i] = coef_g[cbase + i];
  }
  for (int i = 0; i < R_; ++i) {
    float den = 0.f;
    for (int r = 0; r < R_; ++r) den += co[r] * lEtE[r * R_ + i];
    cn[i] = co[i] * num[i] / (den + EPS_);
  }
  if (hi == 0) {
    if (mode <= 1)
      for (int i = 0; i < R_; ++i) coef_g[cbase + i] = cn[i];
    for (int i = 0; i < R_; ++i) lcoef[wave][l * EBP_ + i] = cn[i];
  }

  if (mode <= 1) {
    // coef fragments (A and B for ctc, B for xc); zero-padded -> no masking
    v2f bcf[4];
#pragma unroll
    for (int j = 0; j < 4; ++j) {
      int nr0 = j * 4 + hi * 2;
      bcf[j][0] = lcoef[wave][nr0 * EBP_ + l];
      bcf[j][1] = lcoef[wave][(nr0 + 1) * EBP_ + l];
    }
    // ---- xc[d][r] += x_tile @ coef_new : WMMA, K=n ----
#pragma unroll 2
    for (int db = 0; db < 10; ++db) {
      int dbase = db * 16;
      v8f acc = {};
#pragma unroll
      for (int j = 0; j < 4; ++j) {
        int nn = j * 4 + hi * 2;
        v2f a;
        a[0] = lx[(dbase + l) * PITCH_ + col16 + nn];
        a[1] = lx[(dbase + l) * PITCH_ + col16 + nn + 1];
        acc = __builtin_amdgcn_wmma_f32_16x16x4_f32(false, a, false, bcf[j],
                                                    (short)0, acc, false, false);
      }
      if (l < R_) {
        int dd = dbase + hi * 8;
        for (int i = 0; i < 8; ++i) atomicAdd(&lxc[(dd + i) * R_ + l], acc[i]);
      }
    }
    // ---- ctc = coef^T coef : same fragment both sides ----
    v8f cacc = {};
#pragma unroll
    for (int j = 0; j < 4; ++j)
      cacc = __builtin_amdgcn_wmma_f32_16x16x4_f32(false, bcf[j], false, bcf[j],
                                                   (short)0, cacc, false, false);
    if (lane < R_)
      for (int i = 0; i < R_; ++i) atomicAdd(&lctc[i * R_ + lane], cacc[i]);

    __syncthreads();
    size_t pb = (size_t)(b * NTILES_ + blockIdx.x);
    float* pxcb = pxc + pb * (size_t)(D_ * R_);
    for (int i = tid; i < D_ * R_; i += 256) pxcb[i] = lxc[i];
    if (tid < 64) pctc[pb * 64 + tid] = lctc[tid];
  } else {
    // ---- reconstruction: x_rec = Eb @ coef^T, WMMA, K=r ----
    v2f br[2];
#pragma unroll
    for (int rc = 0; rc < 2; ++rc) {
      int k0 = rc * 4 + hi * 2;
      br[rc][0] = lcoef[wave][l * EBP_ + k0];
      br[rc][1] = lcoef[wave][l * EBP_ + k0 + 1];
    }
#pragma unroll 2
    for (int db = 0; db < 10; ++db) {
      int dbase = db * 16;
      v8f o = {};
#pragma unroll
      for (int rc = 0; rc < 2; ++rc) {
        int k0 = rc * 4 + hi * 2;
        v2f a;
        a[0] = lEb[(dbase + l) * EBP_ + k0];
        a[1] = lEb[(dbase + l) * EBP_ + k0 + 1];
        o = __builtin_amdgcn_wmma_f32_16x16x4_f32(false, a, false, br[rc],
                                                  (short)0, o, false, false);
      }
      size_t obase = (size_t)b * (size_t)(D_ * N_) +
                     (size_t)(dbase + hi * 8) * N_ + tile0 + col16 + l;
      for (int i = 0; i < 8; ++i) out[obase + (size_t)i * N_] = o[i];
    }
  }
}

// ---------------- host orchestration ----------------

extern "C" void kernel_launch(void* const* d_in, const int* in_sizes, int n_in,
                              void* d_out, int out_size, void* d_ws, size_t ws_size,
                              hipStream_t stream) {
  (void)in_sizes; (void)n_in; (void)out_size; (void)ws_size;
  const float* x     = (const float*)d_in[0];
  const float* binit = (const float*)d_in[1];
  float* out = (float*)d_out;
  float* ws  = (float*)d_ws;

  float* E     = ws + O_E;
  float* bases = ws + O_BASES;
  float* Eb    = ws + O_EB;
  float* EtE   = ws + O_ETE;
  float* numb  = ws + O_NUMB;
  float* bc    = ws + O_BC;
  float* t2    = ws + O_T2;
  float* xc    = ws + O_XC;
  float* ctc   = ws + O_CTC;
  float* coef  = ws + O_COEF;
  float* pxc   = ws + O_PXC;
  float* pctc  = ws + O_PCTC;

  k_build_E<<<(D_ * K_ + 255) / 256, 256, 0, stream>>>(E);
  k_init_bases<<<NB_, 256, 0, stream>>>(binit, bases);
  k_Eb<<<dim3(5, NB_), 256, 0, stream>>>(E, bases, Eb);
  k_EtE<<<NB_, 64, 0, stream>>>(Eb, EtE);

  // pass 0: softmax init + step-1 coef update + xc/ctc accumulation
  k_pass<<<dim3(NTILES_, NB_), 256, 0, stream>>>(x, Eb, EtE, coef, pxc, pctc, out, 0);

  for (int s = 0; s < 4; ++s) {   // 4 bases updates (steps 1..4)
    k_reduce<<<dim3(6, NB_), 256, 0, stream>>>(pxc, pctc, xc, ctc);
    k_numb_bc<<<dim3(19, NB_), 256, 0, stream>>>(E, xc, bases, ctc, numb, bc);
    k_t2<<<dim3(5, NB_), 256, 0, stream>>>(E, bc, t2);
    k_denb<<<dim3(19, NB_), 256, 0, stream>>>(E, t2, numb, bases);
    k_Eb<<<dim3(5, NB_), 256, 0, stream>>>(E, bases, Eb);
    k_EtE<<<NB_, 64, 0, stream>>>(Eb, EtE);
    if (s < 3)  // steps 2..4 coef updates + accumulation
      k_pass<<<dim3(NTILES_, NB_), 256, 0, stream>>>(x, Eb, EtE, coef, pxc, pctc, out, 1);
  }

  // final: compute_coef + reconstruction
  k_pass<<<dim3(NTILES_, NB_), 256, 0, stream>>>(x, Eb, EtE, coef, pxc, pctc, out, 2);
}